// Attention_49211735277611
// MI455X (gfx1250) — compile-verified
//
#include <hip/hip_runtime.h>
#include <hip/hip_bf16.h>
#include <float.h>

typedef __attribute__((ext_vector_type(2))) float v2f;
typedef __attribute__((ext_vector_type(8))) float v8f;
typedef __attribute__((ext_vector_type(4))) unsigned int v4u;
typedef __attribute__((ext_vector_type(8))) int v8i;
typedef __attribute__((ext_vector_type(4))) int v4i;

// Problem constants (match reference)
constexpr int Bc = 2, Nc = 2048, DIMc = 512, Hc = 8, DHc = 64, INNERc = 512;
constexpr int QKVc = 3 * INNERc;   // 1536
constexpr float SCALEc = 0.125f;   // 64^-0.5

constexpr int KSTRIDE = 66;        // 64 data dwords + 2 pad dwords (TDM padding)

// D = A(16x4 f32) x B(4x16 f32) + C(16x16 f32), full fp32 precision.
__device__ __forceinline__ v8f wmma_f32(v2f a, v2f b, v8f c) {
    return __builtin_amdgcn_wmma_f32_16x16x4_f32(
        /*neg_a=*/false, a, /*neg_b=*/false, b,
        /*c_mod=*/(short)0, c, /*reuse_a=*/false, /*reuse_b=*/false);
}

// ---------------------------------------------------------------------------
// Tensor Data Mover: DMA a 16x64 f32 tile (rows of K or V) from global to LDS.
// LDS rows are padded 64 -> 66 dwords via D# pad_interval/pad_amount to avoid
// bank conflicts on the strided B-operand reads.
// ---------------------------------------------------------------------------
__device__ __forceinline__ void tdm_tile_load(const float* gsrc, float* lds_dst) {
    const unsigned long long ga = (unsigned long long)(uintptr_t)gsrc;
    v4u g0;
    g0[0] = 1u;                                   // count=1, user mode
    g0[1] = (unsigned)(uintptr_t)lds_dst;         // LDS byte address (low 32b)
    g0[2] = (unsigned)ga;                         // global_addr[31:0]
    g0[3] = (unsigned)((ga >> 32) & 0x1FFFFFFull) // global_addr[56:32]
            | (2u << 30);                         // type=2 ("image")
    v8i g1;
    g1[0] = (2 << 16)      // data_size = 4 bytes
          | (1 << 20)      // pad_enable
          | (5 << 22)      // pad_interval: 64 dwords
          | (1 << 25);     // pad_amount: 2 dwords
    g1[1] = (int)(64u << 16);    // tensor_dim0 = 64 (bits 79:48, low half)
    g1[2] = (int)(2048u << 16);  // tensor_dim0 hi=0 | tensor_dim1 = 2048 (lo16)
    g1[3] = (int)(64u << 16);    // tensor_dim1 hi=0 | tile_dim0 = 64
    g1[4] = 16;                  // tile_dim1 = 16, tile_dim2 = 0
    g1[5] = 64;                  // tensor_dim0_stride = 64
    g1[6] = 0;                   // stride hi | tensor_dim1_stride lo
    g1[7] = 0;
    v4i z4 = {0, 0, 0, 0};
#if defined(__clang_major__) && __clang_major__ >= 23
    v8i z8 = {0, 0, 0, 0, 0, 0, 0, 0};
    __builtin_amdgcn_tensor_load_to_lds(g0, g1, z4, z4, z8, 0);
#else
    __builtin_amdgcn_tensor_load_to_lds(g0, g1, z4, z4, 0);
#endif
}

// ---------------------------------------------------------------------------
// Kernel 1: qkv = x @ Wqkv ; scatter into Q (pre-scaled), K, V as [B,H,N,DH]
// One wave computes one 16x16 output tile. Mtiles=256 (B*N/16), Ntiles=96.
// ---------------------------------------------------------------------------
__global__ __launch_bounds__(128) void qkv_kernel(const float* __restrict__ x,
                                                  const float* __restrict__ Wqkv,
                                                  float* __restrict__ qo,
                                                  float* __restrict__ ko,
                                                  float* __restrict__ vo) {
    const int wave  = blockIdx.x * 4 + (threadIdx.x >> 5);
    const int lane  = threadIdx.x & 31;
    const int mt    = wave / 96;            // row tile of [4096]
    const int nt    = wave % 96;            // col tile of [1536]
    const int row16 = lane & 15;
    const int koff  = (lane >> 4) << 1;     // lanes 0-15 -> K {0,1}, 16-31 -> {2,3}
    const int col   = nt * 16 + row16;

    v8f acc;
#pragma unroll
    for (int r = 0; r < 8; ++r) acc[r] = 0.0f;

    const float* xrow = x + (size_t)(mt * 16 + row16) * DIMc;
    for (int kk = 0; kk < DIMc; kk += 4) {
        v2f a = *(const v2f*)(xrow + kk + koff);
        v2f b;
        b.x = Wqkv[(size_t)(kk + koff) * QKVc + col];
        b.y = Wqkv[(size_t)(kk + koff + 1) * QKVc + col];
        acc = wmma_f32(a, b, acc);
    }

    const int part   = nt / 32;                 // 0=q 1=k 2=v
    const int within = (nt * 16) % INNERc;
    const int h      = within / DHc;
    const int d      = (within % DHc) + row16;
    float* dst       = (part == 0) ? qo : (part == 1) ? ko : vo;
    const float scl  = (part == 0) ? SCALEc : 1.0f;
    const int rhalf  = (lane & 16) ? 8 : 0;

#pragma unroll
    for (int r = 0; r < 8; ++r) {
        const int grow = mt * 16 + r + rhalf;   // row in [0, B*N)
        const int b_   = grow >> 11;            // / 2048
        const int n    = grow & 2047;
        dst[((size_t)(b_ * Hc + h) * Nc + n) * DHc + d] = acc[r] * scl;
    }
}

// ---------------------------------------------------------------------------
// Kernel 2: two-pass flash attention with exact attn_avg.
// One wave per (b, h, 16-row query tile). grid = B*H*(N/16) = 2048.
// K/V tiles are DMA'd to LDS by the Tensor Data Mover, double-buffered.
// ---------------------------------------------------------------------------
__global__ __launch_bounds__(32) void attn_kernel(const float* __restrict__ qb,
                                                  const float* __restrict__ kb,
                                                  const float* __restrict__ vb,
                                                  const unsigned char* __restrict__ pad,
                                                  const float* __restrict__ bias,
                                                  float* __restrict__ Obuf,
                                                  float* __restrict__ attn_avg) {
    const int wid   = blockIdx.x;
    const int it    = wid & 127;            // query tile (N/16 = 128)
    const int bh    = wid >> 7;
    const int h     = bh % Hc;
    const int b     = bh / Hc;
    const int lane  = threadIdx.x;
    const int row16 = lane & 15;
    const int koff  = (lane >> 4) << 1;
    const int rhalf = (lane & 16) ? 8 : 0;
    const int it16  = it * 16;

    const float* qbase = qb + (size_t)(b * Hc + h) * Nc * DHc;
    const float* kbase = kb + (size_t)(b * Hc + h) * Nc * DHc;
    const float* vbase = vb + (size_t)(b * Hc + h) * Nc * DHc;

    __shared__ float Kbuf[2][16 * KSTRIDE + 16];   // TDM double buffer (K)
    __shared__ float Vbuf[2][16 * KSTRIDE + 16];   // TDM double buffer (V)
    __shared__ float Plds[16][17];                 // P: C-layout -> A-layout

    // Load Q tile (already scaled) in WMMA-A layout: 32 floats per lane.
    float qa[32];
    {
        const float* qrow = qbase + (size_t)(it16 + row16) * DHc;
#pragma unroll
        for (int s = 0; s < 16; ++s) {
            v2f t = *(const v2f*)(qrow + 4 * s + koff);
            qa[2 * s]     = t.x;
            qa[2 * s + 1] = t.y;
        }
    }

    // Score tile for key tile jt; K tile read from LDS (rows padded KSTRIDE).
    auto computeS = [&](int jt, const float* Kl, v8f& S) {
#pragma unroll
        for (int r = 0; r < 8; ++r) S[r] = 0.0f;
        const float* kp = Kl + row16 * KSTRIDE + koff;   // row n=row16 of K tile
#pragma unroll
        for (int s = 0; s < 16; ++s) {
            v2f a; a.x = qa[2 * s]; a.y = qa[2 * s + 1];
            v2f bv = *(const v2f*)(kp + 4 * s);
            S = wmma_f32(a, bv, S);
        }
        const int jcol = jt * 16 + row16;
        const bool padj = pad[b * Nc + jcol] != 0;
#pragma unroll
        for (int r = 0; r < 8; ++r) {
            const int irow = it16 + r + rhalf;
            float sv = S[r];
            if (padj || (jcol > irow)) sv = -FLT_MAX;   // padding / causal mask
            sv += bias[(size_t)(h * Nc + irow) * Nc + jcol];
            S[r] = sv;
        }
    };

    // ---- Pass 1: online row max m[] and sum-of-exp l[] (K only) ----
    float m[8], l[8];
#pragma unroll
    for (int r = 0; r < 8; ++r) { m[r] = -FLT_MAX; l[r] = 0.0f; }

    tdm_tile_load(kbase, Kbuf[0]);
    for (int jt = 0; jt <= it; ++jt) {
        if (jt < it) {
            tdm_tile_load(kbase + (size_t)(jt + 1) * 16 * DHc, Kbuf[(jt + 1) & 1]);
            __builtin_amdgcn_s_wait_tensorcnt(1);   // tile jt is in LDS
        } else {
            __builtin_amdgcn_s_wait_tensorcnt(0);
        }
        asm volatile("" ::: "memory");

        v8f S;
        computeS(jt, Kbuf[jt & 1], S);
#pragma unroll
        for (int r = 0; r < 8; ++r) {
            float rm = S[r];
            rm = fmaxf(rm, __shfl_xor(rm, 1, 16));
            rm = fmaxf(rm, __shfl_xor(rm, 2, 16));
            rm = fmaxf(rm, __shfl_xor(rm, 4, 16));
            rm = fmaxf(rm, __shfl_xor(rm, 8, 16));
            const float mnew = fmaxf(m[r], rm);
            float ex = __expf(S[r] - mnew);
            ex += __shfl_xor(ex, 1, 16);
            ex += __shfl_xor(ex, 2, 16);
            ex += __shfl_xor(ex, 4, 16);
            ex += __shfl_xor(ex, 8, 16);
            l[r] = l[r] * __expf(m[r] - mnew) + ex;
            m[r] = mnew;
        }
    }

    // ---- Pass 2: normalized probs -> attn_avg (atomic) and O = P @ V ----
    v8f o[4];
#pragma unroll
    for (int dt = 0; dt < 4; ++dt)
#pragma unroll
        for (int r = 0; r < 8; ++r) o[dt][r] = 0.0f;

    tdm_tile_load(kbase, Kbuf[0]);
    tdm_tile_load(vbase, Vbuf[0]);
    for (int jt = 0; jt <= it; ++jt) {
        if (jt < it) {
            tdm_tile_load(kbase + (size_t)(jt + 1) * 16 * DHc, Kbuf[(jt + 1) & 1]);
            tdm_tile_load(vbase + (size_t)(jt + 1) * 16 * DHc, Vbuf[(jt + 1) & 1]);
            __builtin_amdgcn_s_wait_tensorcnt(2);   // pair jt is in LDS
        } else {
            __builtin_amdgcn_s_wait_tensorcnt(0);
        }
        asm volatile("" ::: "memory");

        v8f S;
        computeS(jt, Kbuf[jt & 1], S);
        const int jcol = jt * 16 + row16;
#pragma unroll
        for (int r = 0; r < 8; ++r) {
            const int irow = it16 + r + rhalf;
            const float p = __expf(S[r] - m[r]) / l[r];
            S[r] = p;
            atomicAdd(&attn_avg[(size_t)(b * Nc + irow) * Nc + jcol],
                      p * (1.0f / Hc));
        }
        // Stage P (C-layout) into LDS, reload in WMMA-A layout.
#pragma unroll
        for (int r = 0; r < 8; ++r) Plds[r + rhalf][row16] = S[r];
        __syncthreads();

        const float* Vl = Vbuf[jt & 1];
#pragma unroll
        for (int dt = 0; dt < 4; ++dt) {
#pragma unroll
            for (int s = 0; s < 4; ++s) {
                v2f a;
                a.x = Plds[row16][4 * s + koff];
                a.y = Plds[row16][4 * s + koff + 1];
                const float* vr = Vl + (4 * s + koff) * KSTRIDE + dt * 16 + row16;
                v2f bv; bv.x = vr[0]; bv.y = vr[KSTRIDE];
                o[dt] = wmma_f32(a, bv, o[dt]);
            }
        }
        __syncthreads();
    }

    // Store O as [B, N, INNER] (head offset h*64) for the output projection.
#pragma unroll
    for (int dt = 0; dt < 4; ++dt)
#pragma unroll
        for (int r = 0; r < 8; ++r) {
            const int irow = it16 + r + rhalf;
            Obuf[(size_t)(b * Nc + irow) * INNERc + h * DHc + dt * 16 + row16] = o[dt][r];
        }
}

// ---------------------------------------------------------------------------
// Kernel 3: out = O @ Wout + bout.  Mtiles=256, Ntiles=32, one wave per tile.
// ---------------------------------------------------------------------------
__global__ __launch_bounds__(128) void proj_kernel(const float* __restrict__ Obuf,
                                                   const float* __restrict__ Wout,
                                                   const float* __restrict__ bout,
                                                   float* __restrict__ out) {
    const int wave  = blockIdx.x * 4 + (threadIdx.x >> 5);
    const int lane  = threadIdx.x & 31;
    const int mt    = wave >> 5;
    const int nt    = wave & 31;
    const int row16 = lane & 15;
    const int koff  = (lane >> 4) << 1;
    const int col   = nt * 16 + row16;

    v8f acc;
#pragma unroll
    for (int r = 0; r < 8; ++r) acc[r] = 0.0f;

    const float* orow = Obuf + (size_t)(mt * 16 + row16) * INNERc;
    for (int kk = 0; kk < INNERc; kk += 4) {
        v2f a = *(const v2f*)(orow + kk + koff);
        v2f b;
        b.x = Wout[(size_t)(kk + koff) * DIMc + col];
        b.y = Wout[(size_t)(kk + koff + 1) * DIMc + col];
        acc = wmma_f32(a, b, acc);
    }

    const float bc   = bout[col];
    const int rbase  = mt * 16 + ((lane & 16) ? 8 : 0);
#pragma unroll
    for (int r = 0; r < 8; ++r)
        out[(size_t)(rbase + r) * DIMc + col] = acc[r] + bc;
}

// ---------------------------------------------------------------------------
extern "C" void kernel_launch(void* const* d_in, const int* in_sizes, int n_in,
                              void* d_out, int out_size, void* d_ws, size_t ws_size,
                              hipStream_t stream) {
    (void)in_sizes; (void)n_in; (void)out_size; (void)ws_size;

    const float*         x    = (const float*)d_in[0];
    const unsigned char* pad  = (const unsigned char*)d_in[1]; // bool mask
    /* d_in[2] = causal mask: recomputed analytically (triu k=1) */
    const float*         bias = (const float*)d_in[3];
    const float*         Wqkv = (const float*)d_in[4];
    const float*         Wout = (const float*)d_in[5];
    const float*         bout = (const float*)d_in[6];

    float* out      = (float*)d_out;
    float* attn_avg = out + (size_t)Bc * Nc * DIMc;

    // Workspace: Q | K | V | O, each B*H*N*DH = 2,097,152 f32 (8 MB)
    const size_t per = (size_t)Bc * Hc * Nc * DHc;
    float* qb = (float*)d_ws;
    float* kb = qb + per;
    float* vb = kb + per;
    float* Ob = vb + per;

    // attn_avg is accumulated with float atomics across heads -> zero it.
    hipMemsetAsync(attn_avg, 0, (size_t)Bc * Nc * Nc * sizeof(float), stream);

    // 1) QKV projection: 256*96 = 24576 wave-tiles, 4 waves per block.
    qkv_kernel<<<6144, 128, 0, stream>>>(x, Wqkv, qb, kb, vb);

    // 2) Attention: one wave per (b, h, 16-row tile), TDM-fed K/V tiles.
    attn_kernel<<<Bc * Hc * (Nc / 16), 32, 0, stream>>>(qb, kb, vb, pad, bias,
                                                        Ob, attn_avg);

    // 3) Output projection: 256*32 = 8192 wave-tiles, 4 waves per block.
    proj_kernel<<<2048, 128, 0, stream>>>(Ob, Wout, bout, out);
}